// VectorQuantizer2_78176994722626
// MI455X (gfx1250) — compile-verified
//
#include <hip/hip_runtime.h>
#include <hip/hip_bf16.h>
#include <math.h>

typedef __attribute__((ext_vector_type(2))) float v2f;
typedef __attribute__((ext_vector_type(8))) float v8f;

#define N_VEC   16384   // rows of z_flat
#define N_E     16384   // codebook entries
#define E_DIM   256
#define TILES   4                       // M-tiles per wave (B-fragment reuse)
#define ROWS_PER_BLOCK (TILES * 16)     // 64
#define A_STRIDE 260                    // conflict-free LDS row stride (floats)

// ---------------------------------------------------------------------------
// Kernel 0: embsq[k] = ||emb_k||^2 ; zero counts ; zero loss accumulator.
// ---------------------------------------------------------------------------
__global__ __launch_bounds__(256) void vq_prep_kernel(
    const float* __restrict__ emb, float* __restrict__ embsq,
    int* __restrict__ counts, float* __restrict__ loss_acc)
{
    const int k = blockIdx.x * 256 + threadIdx.x;
    counts[k] = 0;
    if (k == 0) *loss_acc = 0.0f;
    const float* row = emb + (size_t)k * E_DIM;
    float s = 0.0f;
    for (int d = 0; d < E_DIM; d += 4) {
        float4 v = *(const float4*)(row + d);
        s += v.x * v.x + v.y * v.y + v.z * v.z + v.w * v.w;
    }
    embsq[k] = s;
}

// ---------------------------------------------------------------------------
// Kernel 1: fused distance-GEMM + argmin via V_WMMA_F32_16X16X4_F32.
// Block = 256 threads = 8 waves; block owns 64 z-rows (staged in dynamic LDS).
// Each wave sweeps its 16-code stripe of the whole codebook; every B fragment
// (global) feeds 4 WMMAs against 4 row-tiles (A from LDS). The K-loop is
// software-pipelined one step deep so the WMMAs of step kk overlap the
// LDS/global loads of step kk+1 (no s_wait_*cnt 0x0 in front of each WMMA).
// ---------------------------------------------------------------------------
__global__ __launch_bounds__(256) void vq_argmin_kernel(
    const float* __restrict__ z, const float* __restrict__ emb,
    const float* __restrict__ embsq, int* __restrict__ out_idx)
{
    extern __shared__ float lds_z[];            // ROWS_PER_BLOCK * A_STRIDE floats
    __shared__ float red_val[8][ROWS_PER_BLOCK];
    __shared__ int   red_idx[8][ROWS_PER_BLOCK];

    const int tid  = threadIdx.x;
    const int wave = tid >> 5;
    const int lane = tid & 31;
    const int half = lane >> 4;     // 0: K pair (0,1) / C rows j ; 1: (2,3) / C rows j+8
    const int l16  = lane & 15;
    const int row_base = blockIdx.x * ROWS_PER_BLOCK;

    // Stage the 64x256 z tile into LDS (float4, coalesced).
    for (int i = tid; i < ROWS_PER_BLOCK * (E_DIM / 4); i += 256) {
        const int r = i >> 6;             // row (0..63)
        const int c = (i & 63) << 2;      // col (0..252 step 4)
        float4 v = *(const float4*)(z + (size_t)(row_base + r) * E_DIM + c);
        float* dst = &lds_z[r * A_STRIDE + c];
        dst[0] = v.x; dst[1] = v.y; dst[2] = v.z; dst[3] = v.w;
    }
    __syncthreads();

    // Per-lane A-fragment stream bases (hoisted out of all loops).
    const float* abase[TILES];
#pragma unroll
    for (int t = 0; t < TILES; ++t)
        abase[t] = &lds_z[(t * 16 + l16) * A_STRIDE + half * 2];

    float bestval[TILES][8];
    int   bestidx[TILES][8];
#pragma unroll
    for (int t = 0; t < TILES; ++t)
#pragma unroll
        for (int j = 0; j < 8; ++j) { bestval[t][j] = 3.4e38f; bestidx[t][j] = 0; }

    for (int iter = 0; iter < N_E / 128; ++iter) {
        const int kbase = iter * 128 + wave * 16;           // this wave's 16 codes
        // B fragment: emb row (kbase+l16), dims (kk*4 + half*2 + {0,1}).
        const float* brow = emb + (size_t)(kbase + l16) * E_DIM + half * 2;
        // Prefetch next iteration's code stripe into cache (global_prefetch_b8).
        __builtin_prefetch(brow + 128 * E_DIM, 0, 0);

        v8f acc[TILES];
#pragma unroll
        for (int t = 0; t < TILES; ++t) { v8f zero = {}; acc[t] = zero; }

        // ---- software-pipelined K loop (depth 1) ----
        v2f a_cur[TILES], b_cur;
        b_cur.x = brow[0]; b_cur.y = brow[1];
#pragma unroll
        for (int t = 0; t < TILES; ++t) {
            a_cur[t].x = abase[t][0]; a_cur[t].y = abase[t][1];
        }

        for (int kk = 0; kk < E_DIM / 4 - 1; ++kk) {
            // Issue next step's loads first ...
            v2f b_nxt;
            b_nxt.x = brow[kk * 4 + 4]; b_nxt.y = brow[kk * 4 + 5];
            v2f a_nxt[TILES];
#pragma unroll
            for (int t = 0; t < TILES; ++t) {
                a_nxt[t].x = abase[t][kk * 4 + 4];
                a_nxt[t].y = abase[t][kk * 4 + 5];
            }
            // ... then consume the previous step's fragments.
#pragma unroll
            for (int t = 0; t < TILES; ++t)
                acc[t] = __builtin_amdgcn_wmma_f32_16x16x4_f32(
                    false, a_cur[t], false, b_cur,
                    (short)0, acc[t], false, false);
#pragma unroll
            for (int t = 0; t < TILES; ++t) a_cur[t] = a_nxt[t];
            b_cur = b_nxt;
        }
        // Peeled final step (no further loads -> no OOB read past last emb row).
#pragma unroll
        for (int t = 0; t < TILES; ++t)
            acc[t] = __builtin_amdgcn_wmma_f32_16x16x4_f32(
                false, a_cur[t], false, b_cur,
                (short)0, acc[t], false, false);

        // C layout: acc[t][j] = S[row = t*16 + j + 8*half][code = kbase + l16].
        const int   code = kbase + l16;
        const float en   = embsq[code];
#pragma unroll
        for (int t = 0; t < TILES; ++t) {
#pragma unroll
            for (int j = 0; j < 8; ++j) {
                const float d = en - 2.0f * acc[t][j];  // |z|^2 dropped (row-constant)
                if (d < bestval[t][j]) { bestval[t][j] = d; bestidx[t][j] = code; }
            }
        }
    }

    // Reduce across the 16 lanes of each half (xor 8,4,2,1 stays within half).
#pragma unroll
    for (int t = 0; t < TILES; ++t) {
#pragma unroll
        for (int j = 0; j < 8; ++j) {
            float v = bestval[t][j]; int ix = bestidx[t][j];
            for (int off = 8; off > 0; off >>= 1) {
                float ov = __shfl_xor(v, off, 32);
                int   oi = __shfl_xor(ix, off, 32);
                if (ov < v || (ov == v && oi < ix)) { v = ov; ix = oi; }
            }
            if (l16 == 0) {
                red_val[wave][t * 16 + j + 8 * half] = v;
                red_idx[wave][t * 16 + j + 8 * half] = ix;
            }
        }
    }
    __syncthreads();

    // Final reduce over the 8 waves (each handled disjoint code stripes).
    if (tid < ROWS_PER_BLOCK) {
        float v = red_val[0][tid]; int ix = red_idx[0][tid];
        for (int w = 1; w < 8; ++w) {
            float ov = red_val[w][tid]; int oi = red_idx[w][tid];
            if (ov < v || (ov == v && oi < ix)) { v = ov; ix = oi; }
        }
        out_idx[row_base + tid] = ix;
    }
}

// ---------------------------------------------------------------------------
// Kernel 2: gather z_q = emb[idx], emit idx as float, accumulate sum((zq-z)^2)
// and the histogram. One block (256 threads) per row.
// ---------------------------------------------------------------------------
__global__ __launch_bounds__(256) void vq_gather_kernel(
    const float* __restrict__ z, const float* __restrict__ emb,
    const int* __restrict__ idx, float* __restrict__ zq_out,
    float* __restrict__ idx_out, float* __restrict__ loss_acc,
    int* __restrict__ counts)
{
    __shared__ float red[256];
    const int n = blockIdx.x;
    const int d = threadIdx.x;
    const int k = idx[n];

    const float e  = emb[(size_t)k * E_DIM + d];
    const float zz = z[(size_t)n * E_DIM + d];
    zq_out[(size_t)n * E_DIM + d] = e;

    const float diff = e - zz;
    red[d] = diff * diff;
    __syncthreads();
    for (int s = 128; s > 0; s >>= 1) {
        if (d < s) red[d] += red[d + s];
        __syncthreads();
    }
    if (d == 0) {
        atomicAdd(loss_acc, red[0]);
        atomicAdd(&counts[k], 1);
        idx_out[n] = (float)k;
    }
}

// ---------------------------------------------------------------------------
// Kernel 3: scalars — loss = 1.25 * sumsq / (N*D); perplexity from counts.
// ---------------------------------------------------------------------------
__global__ __launch_bounds__(256) void vq_final_kernel(
    const float* __restrict__ loss_acc, const int* __restrict__ counts,
    float* __restrict__ loss_out, float* __restrict__ perp_out)
{
    __shared__ float red[256];
    const int t = threadIdx.x;
    float local = 0.0f;
    for (int k = t; k < N_E; k += 256) {
        const float p = (float)counts[k] * (1.0f / (float)N_VEC);
        local += p * logf(p + 1e-12f);
    }
    red[t] = local;
    __syncthreads();
    for (int s = 128; s > 0; s >>= 1) {
        if (t < s) red[t] += red[t + s];
        __syncthreads();
    }
    if (t == 0) {
        *perp_out = expf(-red[0]);
        *loss_out = 1.25f * (*loss_acc) * (1.0f / ((float)N_VEC * (float)E_DIM));
    }
}

// ---------------------------------------------------------------------------
extern "C" void kernel_launch(void* const* d_in, const int* in_sizes, int n_in,
                              void* d_out, int out_size, void* d_ws, size_t ws_size,
                              hipStream_t stream)
{
    const float* z   = (const float*)d_in[0];   // [16,1024,256]
    const float* emb = (const float*)d_in[1];   // [16384,256]

    // Workspace layout (all 4-byte elements):
    int*   idx    = (int*)d_ws;                      // [16384]
    int*   counts = (int*)d_ws + N_VEC;              // [16384]
    float* embsq  = (float*)d_ws + 2 * N_VEC;        // [16384]
    float* lacc   = (float*)d_ws + 3 * N_VEC;        // [1]

    // Output layout: z_q | loss | idx(as float) | perplexity
    float* zq_out   = (float*)d_out;
    float* loss_out = zq_out + (size_t)N_VEC * E_DIM;
    float* idx_out  = loss_out + 1;
    float* perp_out = idx_out + N_VEC;

    const size_t lds_bytes = (size_t)ROWS_PER_BLOCK * A_STRIDE * sizeof(float);

    vq_prep_kernel  <<<N_E / 256, 256, 0, stream>>>(emb, embsq, counts, lacc);
    vq_argmin_kernel<<<N_VEC / ROWS_PER_BLOCK, 256, lds_bytes, stream>>>(z, emb, embsq, idx);
    vq_gather_kernel<<<N_VEC, 256, 0, stream>>>(z, emb, idx, zq_out, idx_out, lacc, counts);
    vq_final_kernel <<<1, 256, 0, stream>>>(lacc, counts, loss_out, perp_out);
}